// ScoreNet_40965398069593
// MI455X (gfx1250) — compile-verified
//
#include <hip/hip_runtime.h>
#include <hip/hip_bf16.h>
#include <stdint.h>
#include <math.h>

// ---------------------------------------------------------------------------
// CDNA5 async global<->LDS helpers (ASYNCcnt-tracked data movement)
// ---------------------------------------------------------------------------
__device__ __forceinline__ void async_load_b128(uint32_t lds_off, const void* gaddr) {
    // LDS target at byte offset lds_off ; global addr from 64-bit VGPR pair (GV mode)
    asm volatile("global_load_async_to_lds_b128 %0, %1, off th:TH_LOAD_NT"
                 :: "v"(lds_off), "v"(gaddr) : "memory");
}

__device__ __forceinline__ void async_store_b128(void* gaddr, uint32_t lds_off) {
    asm volatile("global_store_async_from_lds_b128 %0, %1, off th:TH_STORE_NT"
                 :: "v"(gaddr), "v"(lds_off) : "memory");
}

__device__ __forceinline__ void wait_async0() {
    asm volatile("s_wait_asynccnt 0x0" ::: "memory");
}

// ---------------------------------------------------------------------------
// Kernel 1: scores = attn · normalized(conv_weight); top-26 per batch.
// One block per batch (16 blocks), 256 threads (one per patch index).
// Tie-break matches jax.lax.top_k: larger value first, equal -> lower index.
// ---------------------------------------------------------------------------
__global__ void topk_scores_kernel(const float* __restrict__ attn,   // [16,3,256]
                                   const float* __restrict__ cwraw,  // [3]
                                   int* __restrict__ idx_out)        // [16,26]
{
    const int b = blockIdx.x;
    const int t = threadIdx.x;   // 0..255 = patch index n

    // min-max normalize then sum-normalize the 3 conv weights
    float w0 = cwraw[0], w1 = cwraw[1], w2 = cwraw[2];
    float wmin = fminf(w0, fminf(w1, w2));
    float wmax = fmaxf(w0, fmaxf(w1, w2));
    float inv  = 1.0f / (wmax - wmin);
    float c0 = (w0 - wmin) * inv;
    float c1 = (w1 - wmin) * inv;
    float c2 = (w2 - wmin) * inv;
    float s  = c0 + c1 + c2;
    c0 /= s; c1 /= s; c2 /= s;

    const float* a = attn + (size_t)b * 768;
    float score = a[t] * c0 + a[256 + t] * c1 + a[512 + t] * c2;

    __shared__ float sc[256];
    __shared__ float rv[256];
    __shared__ int   ri[256];
    sc[t] = score;
    __syncthreads();

    for (int j = 0; j < 26; ++j) {
        rv[t] = sc[t];
        ri[t] = t;
        __syncthreads();
        for (int stride = 128; stride > 0; stride >>= 1) {
            if (t < stride) {
                float v1 = rv[t], v2 = rv[t + stride];
                int   i1 = ri[t], i2 = ri[t + stride];
                if (v2 > v1 || (v2 == v1 && i2 < i1)) { rv[t] = v2; ri[t] = i2; }
            }
            __syncthreads();
        }
        if (t == 0) {
            int best = ri[0];
            idx_out[b * 26 + j] = best;
            sc[best] = -INFINITY;   // remove winner for next round
        }
        __syncthreads();
    }
}

// ---------------------------------------------------------------------------
// Kernel 2: gather selected 64x64x3 patches via async global->LDS->global.
// One block per (b, j): 16*26 = 416 blocks, 256 threads, 48 KB LDS staging.
// Each thread moves 12 float4 (B128) elements; it stores exactly the LDS
// bytes its own async loads wrote, so only s_wait_asynccnt is needed
// (ASYNCcnt is per-wave; completion implies the LDS write has landed).
// ---------------------------------------------------------------------------
__global__ void gather_patches_kernel(const float* __restrict__ x,  // [16,3,1024,1024]
                                      const int*   __restrict__ idx,// [16,26]
                                      float*       __restrict__ out)// [16,26,3,64,64]
{
    const int blk = blockIdx.x;        // 0..415
    const int b   = blk / 26;
    const int j   = blk % 26;
    const int t   = threadIdx.x;       // 0..255

    const int p  = idx[b * 26 + j];    // patch index 0..255
    const int mi = p >> 4;             // patch row  (m  = 16)
    const int wi = p & 15;             // patch col  (nw = 16)

    __shared__ float4 buf[3072];       // 3 ch * 64 rows * 16 float4 = 48 KB
    const uint32_t lds_base = (uint32_t)(uintptr_t)(void*)&buf[0];

    // source: x[b, c, mi*64 + ph, wi*64 + 4*q ...]
    const float* xb = x + (size_t)b * 3u * 1048576u
                        + (size_t)mi * 64u * 1024u
                        + (size_t)wi * 64u;
    // dest: contiguous 12288 floats per (b,j)
    float* ob = out + (size_t)(b * 26 + j) * 12288u;

    #pragma unroll
    for (int u = 0; u < 12; ++u) {
        const int i  = u * 256 + t;    // float4 index 0..3071 (c-major, row, col)
        const int c  = i >> 10;        // / 1024
        const int r  = i & 1023;
        const int ph = r >> 4;         // row within patch
        const int q  = r & 15;         // float4 within row
        const float* src = xb + (size_t)c * 1048576u + (size_t)ph * 1024u + (size_t)q * 4u;
        async_load_b128(lds_base + (uint32_t)i * 16u, src);
    }

    wait_async0();                     // all LDS writes from our loads landed

    #pragma unroll
    for (int u = 0; u < 12; ++u) {
        const int i = u * 256 + t;
        float* dst = ob + (size_t)i * 4u;
        async_store_b128(dst, lds_base + (uint32_t)i * 16u);
    }

    wait_async0();                     // drain before LDS/wave teardown
}

// ---------------------------------------------------------------------------
// Launch
// ---------------------------------------------------------------------------
extern "C" void kernel_launch(void* const* d_in, const int* in_sizes, int n_in,
                              void* d_out, int out_size, void* d_ws, size_t ws_size,
                              hipStream_t stream) {
    const float* x    = (const float*)d_in[0];  // [16,3,1024,1024]
    const float* attn = (const float*)d_in[1];  // [16,3,256]
    const float* cw   = (const float*)d_in[2];  // [1,3,1,1]
    // d_in[3] = noise: only affects gradients; forward output == hard gather.
    // d_in[4] = use_hard_topk: forward value identical either way.

    int* idx = (int*)d_ws;                      // 16*26 ints

    topk_scores_kernel<<<16, 256, 0, stream>>>(attn, cw, idx);
    gather_patches_kernel<<<416, 256, 0, stream>>>(x, idx, (float*)d_out);
}